// LightGCN_9861244912155
// MI455X (gfx1250) — compile-verified
//
#include <hip/hip_runtime.h>

typedef __attribute__((ext_vector_type(2))) float v2f;
typedef __attribute__((ext_vector_type(8))) float v8f;

#define NODES 100000
#define DIM 64

// ---------------------------------------------------------------- utilities
__global__ void k_zero(float* __restrict__ p, long n) {
  long i = (long)blockIdx.x * blockDim.x + threadIdx.x;
  long s = (long)gridDim.x * blockDim.x;
  for (; i < n; i += s) p[i] = 0.0f;
}

// deg[col[e]] += 1
__global__ void k_deg(const int* __restrict__ col, float* __restrict__ deg, int E) {
  int e = blockIdx.x * blockDim.x + threadIdx.x;
  if (e < E) {
    __hip_atomic_fetch_add(&deg[col[e]], 1.0f, __ATOMIC_RELAXED,
                           __HIP_MEMORY_SCOPE_AGENT);
  }
}

// deg -> dis in place:  d>0 ? rsqrt(max(d,1)) : 0
__global__ void k_dis(float* __restrict__ deg, int N) {
  int i = blockIdx.x * blockDim.x + threadIdx.x;
  if (i < N) {
    float d = deg[i];
    deg[i] = d > 0.0f ? rsqrtf(fmaxf(d, 1.0f)) : 0.0f;
  }
}

// xn[col] += (dis[row]*dis[col]) * x[row]   — 16 threads per edge, float4 each
__global__ void k_prop(const int* __restrict__ row, const int* __restrict__ col,
                       const float* __restrict__ dis, const float* __restrict__ x,
                       float* __restrict__ xn, int E) {
  long t = (long)blockIdx.x * blockDim.x + threadIdx.x;
  int e = (int)(t >> 4);
  if (e >= E) return;
  int j = (int)(t & 15);
  int r = row[e], c = col[e];
  float w = dis[r] * dis[c];
  const float4 v = *((const float4*)(x + (long)r * DIM) + j);
  float* dst = xn + (long)c * DIM + (j << 2);
  __hip_atomic_fetch_add(dst + 0, w * v.x, __ATOMIC_RELAXED, __HIP_MEMORY_SCOPE_AGENT);
  __hip_atomic_fetch_add(dst + 1, w * v.y, __ATOMIC_RELAXED, __HIP_MEMORY_SCOPE_AGENT);
  __hip_atomic_fetch_add(dst + 2, w * v.z, __ATOMIC_RELAXED, __HIP_MEMORY_SCOPE_AGENT);
  __hip_atomic_fetch_add(dst + 3, w * v.w, __ATOMIC_RELAXED, __HIP_MEMORY_SCOPE_AGENT);
}

// out += alpha[ai] * x
__global__ void k_axpy(float* __restrict__ out, const float* __restrict__ x,
                       const float* __restrict__ alpha, int ai, long n) {
  float a = alpha[ai];
  long i = (long)blockIdx.x * blockDim.x + threadIdx.x;
  long s = (long)gridDim.x * blockDim.x;
  for (; i < n; i += s) out[i] += a * x[i];
}

// -------------------------------------------------- WMMA edge scoring
// One wave32 handles 16 edges. A = out[row[m]] (16x64), B = out[col[n]] (64x16,
// fed as K-major fragments), accumulate C (16x16 f32) with 16 chained
// V_WMMA_F32_16X16X4_F32; result for edge m is the diagonal C[m][m].
//
// f32 16x4 A layout (ISA 7.12.2): lane<16 -> {K=0,K=1} for M=lane;
// lane>=16 -> {K=2,K=3} for M=lane-16. B mirrors with N=lane&15.
// C/D layout: VGPR j, lanes 0-15: (M=j, N=lane); lanes 16-31: (M=j+8, N=lane-16).
__global__ void k_edge_dot(const int* __restrict__ row, const int* __restrict__ col,
                           const float* __restrict__ outf, float* __restrict__ res,
                           int E) {
  int lane = threadIdx.x & 31;
  int wave = (int)(((long)blockIdx.x * blockDim.x + threadIdx.x) >> 5);
  int ebase = wave * 16;
  if (ebase >= E) return;                  // wave-uniform: EXEC stays all-ones

  int m  = lane & 15;
  int kb = (lane >> 4) << 1;               // 0 or 2
  int ea = ebase + m;
  if (ea >= E) ea = E - 1;                 // clamp loads, keep EXEC full

  const float* pa = outf + (long)row[ea] * DIM + kb;   // A fragment source
  const float* pb = outf + (long)col[ea] * DIM + kb;   // B fragment source

  v8f acc = {};
#pragma unroll
  for (int k0 = 0; k0 < DIM; k0 += 4) {
    v2f a = *(const v2f*)(pa + k0);
    v2f b = *(const v2f*)(pb + k0);
    acc = __builtin_amdgcn_wmma_f32_16x16x4_f32(
        /*neg_a=*/false, a, /*neg_b=*/false, b,
        /*c_mod=*/(short)0, acc, /*reuse_a=*/false, /*reuse_b=*/false);
  }

  // diagonal extraction
  int sel, eo;
  if (lane < 8)        { sel = lane;      eo = ebase + lane; }
  else if (lane >= 24) { sel = lane - 24; eo = ebase + lane - 16; }
  else return;

  float r = acc[0];
  r = sel == 1 ? acc[1] : r;
  r = sel == 2 ? acc[2] : r;
  r = sel == 3 ? acc[3] : r;
  r = sel == 4 ? acc[4] : r;
  r = sel == 5 ? acc[5] : r;
  r = sel == 6 ? acc[6] : r;
  r = sel == 7 ? acc[7] : r;
  if (eo < E) res[eo] = r;
}

// ---------------------------------------------------------------- launcher
extern "C" void kernel_launch(void* const* d_in, const int* in_sizes, int n_in,
                              void* d_out, int out_size, void* d_ws, size_t ws_size,
                              hipStream_t stream) {
  const int*   edge  = (const int*)d_in[0];     // [2, E]
  const float* emb   = (const float*)d_in[1];   // [N, 64]
  const float* alpha = (const float*)d_in[2];   // [4]
  int E = in_sizes[0] / 2;
  const int* row = edge;
  const int* col = edge + E;
  float* res = (float*)d_out;

  // workspace layout (floats): dis[N] | xA[N*64] | xB[N*64] | out[N*64]
  float* ws  = (float*)d_ws;
  float* dis = ws;
  float* xA  = ws + NODES;
  float* xB  = xA + (long)NODES * DIM;
  float* out = xB + (long)NODES * DIM;
  long nfeat = (long)NODES * DIM;

  const int B = 256;

  // degree -> normalization
  k_zero<<<512, B, 0, stream>>>(dis, (long)NODES);
  k_deg<<<(E + B - 1) / B, B, 0, stream>>>(col, dis, E);
  k_dis<<<(NODES + B - 1) / B, B, 0, stream>>>(dis, NODES);

  // out = alpha[0] * emb
  k_zero<<<2048, B, 0, stream>>>(out, nfeat);
  k_axpy<<<2048, B, 0, stream>>>(out, emb, alpha, 0, nfeat);

  // 3 propagation layers, ping-pong buffers
  const float* xin = emb;
  float* ping = xA;
  float* pong = xB;
  long pth = (long)E * 16;
  int pgrid = (int)((pth + B - 1) / B);
  for (int i = 0; i < 3; ++i) {
    k_zero<<<2048, B, 0, stream>>>(ping, nfeat);
    k_prop<<<pgrid, B, 0, stream>>>(row, col, dis, xin, ping, E);
    k_axpy<<<2048, B, 0, stream>>>(out, ping, alpha, i + 1, nfeat);
    xin = ping;
    float* t = ping; ping = pong; pong = t;
  }

  // per-edge dot products via WMMA (16 edges per wave32, 8 waves per block)
  int waves  = (E + 15) / 16;
  int blocks = (waves + 7) / 8;
  k_edge_dot<<<blocks, 256, 0, stream>>>(row, col, out, res, E);
}